// NCutsLoss_24240795418782
// MI455X (gfx1250) — compile-verified
//
#include <hip/hip_runtime.h>

// ---------------------------------------------------------------------------
// NCuts loss on MI455X (gfx1250, wave32).
//   W = dissim as [9216 x 9216] f32, S = seg as [16 x 9216] f32.
//   assocA_k = S_k^T W S_k ; assocV_k = S_k^T (W 1) ; out = 16 - sum_k A/V.
// Dominant cost: streaming W once (340 MB @ 23.3 TB/s ~ 14.6 us floor).
// GEMM part (M=9216,N=16,K=9216) rides V_WMMA_F32_16X16X4_F32; degree (row
// sums) accumulates on the VALU (v_pk_add_f32) while the vec4 is live.
// W loads are non-temporal so the streamed 340 MB doesn't evict the L2-hot
// seg working set (576 KB, re-staged by every block).
// ---------------------------------------------------------------------------

typedef __attribute__((ext_vector_type(2))) float v2f;
typedef __attribute__((ext_vector_type(4))) float v4f;
typedef __attribute__((ext_vector_type(8))) float v8f;

#define KCLS 16
#define NPIX 9216                 // 96*96 flattened
#define MN_CHUNKS 16
#define MN_PER_CHUNK (NPIX / MN_CHUNKS)        // 576
#define PAIRS_PER_CHUNK (MN_PER_CHUNK / 2)     // 288
#define WAVES_PER_WG 8
#define ROWS_PER_WAVE 16
#define ROW_GROUPS (NPIX / (ROWS_PER_WAVE * WAVES_PER_WG))  // 72

__global__ __launch_bounds__(256)
void ncuts_main_kernel(const float* __restrict__ seg,
                       const float* __restrict__ W,
                       float* __restrict__ accA,   // [16] global atomics
                       float* __restrict__ accV)   // [16] global atomics
{
    // seg^T chunk staged in LDS, pair-interleaved:
    // lds2[p*16 + k] = { seg[k, base+2p], seg[k, base+2p+1] }
    __shared__ v2f   lds2[PAIRS_PER_CHUNK * KCLS];      // 36 KB
    __shared__ float degLds[WAVES_PER_WG][32];          // 1 KB

    const int tid  = threadIdx.x;
    const int lane = tid & 31;
    const int wv   = tid >> 5;

    const int rg = blockIdx.x >> 4;          // row group [0,72)
    const int ch = blockIdx.x & 15;          // mn chunk  [0,16)
    const int chunk_base = ch * MN_PER_CHUNK;

    // ---- stage seg^T chunk into LDS (vec4 global loads, b64 LDS stores) ----
    // 576*16 floats = 2304 vec4s, 256 threads -> 9 iters. Regular-temporal:
    // seg must stay L2 resident across all 1152 blocks.
    #pragma unroll
    for (int it = 0; it < 9; ++it) {
        int f = it * 256 + tid;              // vec4 index
        int k = f / (MN_PER_CHUNK / 4);      // [0,16)
        int q = f % (MN_PER_CHUNK / 4);      // [0,144)
        v4f s4 = *(const v4f*)(seg + (size_t)k * NPIX + chunk_base + q * 4);
        int p = q * 2;
        lds2[(p    ) * KCLS + k] = v2f{s4.x, s4.y};
        lds2[(p + 1) * KCLS + k] = v2f{s4.z, s4.w};
    }
    __syncthreads();

    // ---- per-wave 16-row tile of W, reduce over this mn chunk -------------
    const int row0 = (rg * WAVES_PER_WG + wv) * ROWS_PER_WAVE;
    const int rrel = lane & 15;              // A-matrix M = lane % 16
    const int hi   = lane >> 4;              // half-wave supplies K-slots 2,3

    // A-fragment source: lane<16 takes mn [m0..m0+3], lane>=16 [m0+4..m0+7];
    // lanes r and r+16 of a row share one 128B line per issue.
    const v4f* wrow = (const v4f*)(W + (size_t)(row0 + rrel) * NPIX
                                     + chunk_base + hi * 4);

    v8f c  = {};                             // Wseg tile [16 rows x 16 k]
    v2f d2 = {0.0f, 0.0f};                   // partial row-sum (degree), packed

    #pragma unroll 4
    for (int s = 0; s < MN_PER_CHUNK / 8; ++s) {   // 72 supersteps, 2 WMMA each
        const v4f w4 = __builtin_nontemporal_load(wrow + s * 2);

        // degree: one v_pk_add_f32 pair, no register shuffling
        const v2f alo = __builtin_shufflevector(w4, w4, 0, 1);
        const v2f ahi = __builtin_shufflevector(w4, w4, 2, 3);
        d2 += alo + ahi;

        const int col = lane & 15;
        const int pb  = s * 4 + (hi << 1);
        const v2f b0 = lds2[(pb    ) * KCLS + col]; // K-slots {0,1}/{2,3} pair 0
        const v2f b1 = lds2[(pb + 1) * KCLS + col]; // K-slots {0,1}/{2,3} pair 1

        // D = A x B + C on the f32 matrix pipe
        c = __builtin_amdgcn_wmma_f32_16x16x4_f32(false, alo, false, b0,
                                                  (short)0, c, false, false);
        c = __builtin_amdgcn_wmma_f32_16x16x4_f32(false, ahi, false, b1,
                                                  (short)0, c, false, false);
    }

    // ---- fold degree halves through LDS -----------------------------------
    degLds[wv][lane] = d2.x + d2.y;
    __syncthreads();

    // C layout: VGPR v, lanes 0-15 -> row v,   N = lane
    //                   lanes 16-31 -> row v+8, N = lane-16
    const int k     = lane & 15;
    const int rbase = hi << 3;
    float pA = 0.0f, pV = 0.0f;
    #pragma unroll
    for (int v = 0; v < 8; ++v) {
        const int r  = rbase + v;
        const float sv = seg[(size_t)k * NPIX + row0 + r];   // L2-hot gather
        const float dg = degLds[wv][r] + degLds[wv][r + 16];
        pA = fmaf(sv, ((const float*)&c)[v], pA);
        pV = fmaf(sv, dg, pV);
    }

    // reduce the two half-waves (same k in lane and lane^16), then atomics
    pA += __shfl_xor(pA, 16, 32);
    pV += __shfl_xor(pV, 16, 32);
    if (lane < 16) {
        atomicAdd(&accA[k], pA);
        atomicAdd(&accV[k], pV);
    }
}

__global__ void ncuts_init_kernel(float* acc)
{
    if (threadIdx.x < 32) acc[threadIdx.x] = 0.0f;
}

__global__ void ncuts_final_kernel(const float* __restrict__ accA,
                                   const float* __restrict__ accV,
                                   float* __restrict__ out)
{
    if (threadIdx.x == 0) {
        float s = 0.0f;
        for (int k = 0; k < KCLS; ++k) s += accA[k] / accV[k];
        out[0] = (float)KCLS - s;
    }
}

extern "C" void kernel_launch(void* const* d_in, const int* in_sizes, int n_in,
                              void* d_out, int out_size, void* d_ws, size_t ws_size,
                              hipStream_t stream) {
    const float* seg    = (const float*)d_in[0];  // [16, 96, 96]
    const float* dissim = (const float*)d_in[1];  // [96, 96, 96, 96]
    // d_in[2] (A) is unused by the reference.

    float* accA = (float*)d_ws;        // [16]
    float* accV = accA + KCLS;         // [16]
    float* out  = (float*)d_out;       // [1]

    ncuts_init_kernel<<<1, 32, 0, stream>>>(accA);
    ncuts_main_kernel<<<dim3(ROW_GROUPS * MN_CHUNKS), dim3(256), 0, stream>>>(
        seg, dissim, accA, accV);
    ncuts_final_kernel<<<1, 32, 0, stream>>>(accA, accV, out);
}